// VectorQuantizer_35029753266253
// MI455X (gfx1250) — compile-verified
//
#include <hip/hip_runtime.h>

typedef __attribute__((ext_vector_type(2))) float v2f;
typedef __attribute__((ext_vector_type(8))) float v8f;

#define TM 128
#define TN 128
#define KC 32
#define LDA 36   // KC + 4 float padding: 144B row stride (16B aligned, spreads banks)
#define LDB 132  // TN + 4 float padding for [k][n] layout in proj kernel

// ---------------------------------------------------------------------------
// CDNA5 async global->LDS copy (ASYNCcnt-tracked, no VGPR staging)
// ---------------------------------------------------------------------------
__device__ __forceinline__ unsigned lds_addr_u32(const void* p) {
  return (unsigned)(uintptr_t)p;  // low 32 bits of generic addr = LDS offset
}

__device__ __forceinline__ void async_copy_b128(unsigned lds_off,
                                                const float* gptr) {
  asm volatile("global_load_async_to_lds_b128 %0, %1, off"
               :
               : "v"(lds_off), "v"(gptr)
               : "memory");
}

__device__ __forceinline__ void wait_async0() {
  asm volatile("s_wait_asynccnt 0x0" ::: "memory");
}

// ---------------------------------------------------------------------------
// 1) per-row rsqrt(max(sum(v^2), eps))
// ---------------------------------------------------------------------------
__global__ __launch_bounds__(256) void row_rnorm_kernel(const float* __restrict__ X,
                                                        float* __restrict__ scale,
                                                        int D, float eps) {
  __shared__ float red[256];
  const int m = blockIdx.x;
  const int tid = threadIdx.x;
  const float* row = X + (size_t)m * D;
  float s = 0.0f;
  for (int i = tid; i < D; i += 256) {
    float v = row[i];
    s += v * v;
  }
  red[tid] = s;
  __syncthreads();
  for (int st = 128; st > 0; st >>= 1) {
    if (tid < st) red[tid] += red[tid + st];
    __syncthreads();
  }
  if (tid == 0) scale[m] = rsqrtf(fmaxf(red[0], eps));
}

// ---------------------------------------------------------------------------
// 2) sim GEMM (f32 WMMA 16x16x4) fused with per-row argmax via packed atomicMax
//    X:[M,K] row-major, W:[N,K] row-major (dot of rows); async double-buffered
// ---------------------------------------------------------------------------
__global__ __launch_bounds__(256) void sim_argmax_kernel(
    const float* __restrict__ X, const float* __restrict__ W,
    const float* __restrict__ qs, const float* __restrict__ ks,
    unsigned long long* __restrict__ packed, int K, int N) {
  __shared__ float As[2][TM * LDA];
  __shared__ float Bs[2][TN * LDA];

  const int tid = threadIdx.x;
  const int lane = tid & 31;
  const int wave = tid >> 5;
  const int l16 = lane & 15;
  const int half = lane >> 4;  // 0: K={0,1}, 1: K={2,3} within a 16x16x4 step
  const int m_base = blockIdx.y * TM;
  const int n_base = blockIdx.x * TN;

  const v8f vzero = {0.f, 0.f, 0.f, 0.f, 0.f, 0.f, 0.f, 0.f};
  v8f acc[8];
#pragma unroll
  for (int t = 0; t < 8; ++t) acc[t] = vzero;

  const int lr = tid >> 1;        // 0..127: tile row this thread copies
  const int lk = (tid & 1) << 4;  // 0 or 16: K-offset this thread copies

  const float* gA = X + (size_t)(m_base + lr) * K + lk;
  const float* gB = W + (size_t)(n_base + lr) * K + lk;
  const unsigned la0 = lds_addr_u32(&As[0][lr * LDA + lk]);
  const unsigned la1 = lds_addr_u32(&As[1][lr * LDA + lk]);
  const unsigned lb0 = lds_addr_u32(&Bs[0][lr * LDA + lk]);
  const unsigned lb1 = lds_addr_u32(&Bs[1][lr * LDA + lk]);

  const int nchunks = K / KC;

  // prologue: issue chunk 0 into buffer 0
#pragma unroll
  for (int q = 0; q < 4; ++q) {
    async_copy_b128(la0 + 16 * q, gA + 4 * q);
    async_copy_b128(lb0 + 16 * q, gB + 4 * q);
  }

  for (int i = 0; i < nchunks; ++i) {
    wait_async0();     // chunk i landed in LDS (my wave's copies)
    __syncthreads();   // all waves' copies landed; prev reads of other buf done

    if (i + 1 < nchunks) {  // prefetch chunk i+1 into the other buffer
      const int kb = (i + 1) * KC;
      const unsigned la = ((i + 1) & 1) ? la1 : la0;
      const unsigned lb = ((i + 1) & 1) ? lb1 : lb0;
#pragma unroll
      for (int q = 0; q < 4; ++q) {
        async_copy_b128(la + 16 * q, gA + kb + 4 * q);
        async_copy_b128(lb + 16 * q, gB + kb + 4 * q);
      }
    }

    const float* Ab = As[i & 1];
    const float* Bb = Bs[i & 1];

    v2f afr[8];
    const float* aw = Ab + (wave * 16 + l16) * LDA + (half << 1);
#pragma unroll
    for (int kk = 0; kk < 8; ++kk) afr[kk] = *(const v2f*)(aw + (kk << 2));

#pragma unroll
    for (int t = 0; t < 8; ++t) {
      const float* bw = Bb + (t * 16 + l16) * LDA + (half << 1);
#pragma unroll
      for (int kk = 0; kk < 8; ++kk) {
        v2f bfr = *(const v2f*)(bw + (kk << 2));
        acc[t] = __builtin_amdgcn_wmma_f32_16x16x4_f32(
            false, afr[kk], false, bfr, (short)0, acc[t], false, false);
      }
    }
  }

  // ---- epilogue: scale by kscale[n], argmax per row, combine via atomicMax ----
  float ksv[8];
#pragma unroll
  for (int t = 0; t < 8; ++t) ksv[t] = ks[n_base + t * 16 + l16];

#pragma unroll
  for (int j = 0; j < 8; ++j) {
    float bv = -3.402823e38f;
    int bn = 0x7FFFFFFF;
#pragma unroll
    for (int t = 0; t < 8; ++t) {
      float v = acc[t][j] * ksv[t];
      int n = n_base + t * 16 + l16;
      if (v > bv || (v == bv && n < bn)) { bv = v; bn = n; }
    }
#pragma unroll
    for (int mk = 1; mk < 16; mk <<= 1) {  // wave32: xor stays within 16-lane half
      float ov = __shfl_xor(bv, mk, 32);
      int on = __shfl_xor(bn, mk, 32);
      if (ov > bv || (ov == bv && on < bn)) { bv = ov; bn = on; }
    }
    if (l16 == 0) {
      int row = m_base + wave * 16 + j + (half << 3);
      float sim = bv * qs[row];
      unsigned int u = __float_as_uint(sim);
      unsigned int key = (u & 0x80000000u) ? ~u : (u | 0x80000000u);
      unsigned long long pk =
          ((unsigned long long)key << 32) |
          (unsigned long long)(~(unsigned int)bn);  // ~idx: ties -> lowest idx
      atomicMax(packed + row, pk);
    }
  }
}

// ---------------------------------------------------------------------------
// 3) unpack argmax, gather codebook row, LayerNorm -> Eln; emit idx & sim
// ---------------------------------------------------------------------------
__global__ __launch_bounds__(256) void gather_ln_kernel(
    const unsigned long long* __restrict__ packed, const float* __restrict__ W,
    const float* __restrict__ gamma, const float* __restrict__ beta,
    int* __restrict__ out_idx, float* __restrict__ out_sim,
    float* __restrict__ Eln, int D) {
  __shared__ float r1[256];
  __shared__ float r2[256];
  __shared__ int s_idx;
  const int m = blockIdx.x;
  const int tid = threadIdx.x;

  if (tid == 0) {
    unsigned long long p = packed[m];
    unsigned int key = (unsigned int)(p >> 32);
    unsigned int u = (key & 0x80000000u) ? (key & 0x7FFFFFFFu) : ~key;
    int idx = (int)(~(unsigned int)(p & 0xFFFFFFFFull));
    out_idx[m] = idx;
    out_sim[m] = __uint_as_float(u);
    s_idx = idx;
  }
  __syncthreads();
  const float* z = W + (size_t)s_idx * D;

  float s1 = 0.0f, s2 = 0.0f;
  for (int i = tid; i < D; i += 256) {
    float v = z[i];
    s1 += v;
    s2 += v * v;
  }
  r1[tid] = s1;
  r2[tid] = s2;
  __syncthreads();
  for (int st = 128; st > 0; st >>= 1) {
    if (tid < st) {
      r1[tid] += r1[tid + st];
      r2[tid] += r2[tid + st];
    }
    __syncthreads();
  }
  const float invD = 1.0f / (float)D;
  float mu = r1[0] * invD;
  float var = fmaxf(r2[0] * invD - mu * mu, 0.0f);
  float rstd = rsqrtf(var + 1e-6f);

  float* orow = Eln + (size_t)m * D;
  for (int i = tid; i < D; i += 256) {
    orow[i] = (z[i] - mu) * rstd * gamma[i] + beta[i];
  }
}

// ---------------------------------------------------------------------------
// 4) output projection: C[M,N] = A[M,K] @ B[K,N]   (f32 WMMA, async pipelined)
//    B tile staged untransposed as [KC][TN+4]; fragments = two strided b32
// ---------------------------------------------------------------------------
__global__ __launch_bounds__(256) void proj_gemm_kernel(
    const float* __restrict__ A, const float* __restrict__ B,
    float* __restrict__ C, int K, int N) {
  __shared__ float As[2][TM * LDA];
  __shared__ float Bs[2][KC * LDB];

  const int tid = threadIdx.x;
  const int lane = tid & 31;
  const int wave = tid >> 5;
  const int l16 = lane & 15;
  const int half = lane >> 4;
  const int m_base = blockIdx.y * TM;
  const int n_base = blockIdx.x * TN;

  const v8f vzero = {0.f, 0.f, 0.f, 0.f, 0.f, 0.f, 0.f, 0.f};
  v8f acc[8];
#pragma unroll
  for (int t = 0; t < 8; ++t) acc[t] = vzero;

  const int lr = tid >> 1;        // A copier: tile row
  const int lk = (tid & 1) << 4;  // A copier: k offset (0|16)
  const int bk = tid >> 3;        // B copier: k row (0..31)
  const int bn = (tid & 7) << 4;  // B copier: n offset (0,16,..,112)

  const float* gA = A + (size_t)(m_base + lr) * K + lk;
  const float* gB = B + (size_t)bk * N + n_base + bn;
  const unsigned la0 = lds_addr_u32(&As[0][lr * LDA + lk]);
  const unsigned la1 = lds_addr_u32(&As[1][lr * LDA + lk]);
  const unsigned lb0 = lds_addr_u32(&Bs[0][bk * LDB + bn]);
  const unsigned lb1 = lds_addr_u32(&Bs[1][bk * LDB + bn]);

  const int nchunks = K / KC;

#pragma unroll
  for (int q = 0; q < 4; ++q) {
    async_copy_b128(la0 + 16 * q, gA + 4 * q);
    async_copy_b128(lb0 + 16 * q, gB + 4 * q);
  }

  for (int i = 0; i < nchunks; ++i) {
    wait_async0();
    __syncthreads();

    if (i + 1 < nchunks) {
      const int kb = (i + 1) * KC;
      const unsigned la = ((i + 1) & 1) ? la1 : la0;
      const unsigned lb = ((i + 1) & 1) ? lb1 : lb0;
#pragma unroll
      for (int q = 0; q < 4; ++q) {
        async_copy_b128(la + 16 * q, gA + kb + 4 * q);
        async_copy_b128(lb + 16 * q, gB + (size_t)kb * N + 4 * q);
      }
    }

    const float* Ab = As[i & 1];
    const float* Bb = Bs[i & 1];

    v2f afr[8];
    const float* aw = Ab + (wave * 16 + l16) * LDA + (half << 1);
#pragma unroll
    for (int kk = 0; kk < 8; ++kk) afr[kk] = *(const v2f*)(aw + (kk << 2));

#pragma unroll
    for (int t = 0; t < 8; ++t) {
#pragma unroll
      for (int kk = 0; kk < 8; ++kk) {
        const float* bw = Bb + ((kk << 2) + (half << 1)) * LDB + t * 16 + l16;
        v2f bfr;
        bfr.x = bw[0];
        bfr.y = bw[LDB];
        acc[t] = __builtin_amdgcn_wmma_f32_16x16x4_f32(
            false, afr[kk], false, bfr, (short)0, acc[t], false, false);
      }
    }
  }

#pragma unroll
  for (int t = 0; t < 8; ++t) {
#pragma unroll
    for (int j = 0; j < 8; ++j) {
      int row = m_base + wave * 16 + j + (half << 3);
      int col = n_base + t * 16 + l16;
      C[(size_t)row * N + col] = acc[t][j];
    }
  }
}

// ---------------------------------------------------------------------------
extern "C" void kernel_launch(void* const* d_in, const int* in_sizes, int n_in,
                              void* d_out, int out_size, void* d_ws,
                              size_t ws_size, hipStream_t stream) {
  (void)n_in;
  (void)out_size;
  (void)ws_size;
  const float* x = (const float*)d_in[0];        // [B,S,D]
  const float* dict_w = (const float*)d_in[1];   // [V,D]
  const float* ln_gamma = (const float*)d_in[2]; // [D]
  const float* ln_beta = (const float*)d_in[3];  // [D]
  const float* o_w = (const float*)d_in[4];      // [D,D]

  const int D = in_sizes[2];      // 1024
  const int V = in_sizes[1] / D;  // 16384
  const int M = in_sizes[0] / D;  // B*S = 8192

  // workspace layout
  char* ws = (char*)d_ws;
  unsigned long long* packed = (unsigned long long*)ws;          // M * 8
  float* qscale = (float*)(ws + (size_t)M * 8);                  // M * 4
  float* kscale = (float*)(ws + (size_t)M * 8 + (size_t)M * 4);  // V * 4
  float* Eln = (float*)(ws + (size_t)M * 8 + (size_t)M * 4 + (size_t)V * 4);

  int* out_idx = (int*)d_out;               // [M] int32
  float* out_sim = (float*)d_out + M;       // [M] f32
  float* out_proj = (float*)d_out + 2 * M;  // [M,D] f32

  row_rnorm_kernel<<<M, 256, 0, stream>>>(x, qscale, D, 1e-12f);
  row_rnorm_kernel<<<V, 256, 0, stream>>>(dict_w, kscale, D, 1e-12f);
  hipMemsetAsync(packed, 0, (size_t)M * 8, stream);

  dim3 g1(V / TN, M / TM);
  sim_argmax_kernel<<<g1, 256, 0, stream>>>(x, dict_w, qscale, kscale, packed,
                                            D, V);

  gather_ln_kernel<<<M, 256, 0, stream>>>(packed, dict_w, ln_gamma, ln_beta,
                                          out_idx, out_sim, Eln, D);

  dim3 g2(D / TN, M / TM);
  proj_gemm_kernel<<<g2, 256, 0, stream>>>(Eln, o_w, out_proj, D, D);
}